// RefineGNN_26431228739692
// MI455X (gfx1250) — compile-verified
//
#include <hip/hip_runtime.h>
#include <hip/hip_bf16.h>

#define N_NODES 50000
#define E_EDGES 600000
#define IN_DIM 64
#define HIDDEN 128
#define LAYERS 4
#define EDGE_DIM 8
#define EPS_NORM 1e-5f

typedef __attribute__((ext_vector_type(2))) float v2f;
typedef __attribute__((ext_vector_type(8))) float v8f;

// ---------------------------------------------------------------------------
// WMMA helpers: D = A(16x4,f32) * B(4x16,f32) + C(16x16,f32), wave32.
// A layout (ISA 7.12.2): lanes 0-15 rows 0-15 hold K={k0,k0+1}, lanes 16-31
// hold K={k0+2,k0+3}.  B mirrors it across rows; C: VGPR i holds row i (lanes
// 0-15) / row 8+i (lanes 16-31), col = lane&15.
// ---------------------------------------------------------------------------
__device__ __forceinline__ v8f wmma_k4(v2f a, v2f b, v8f c) {
    return __builtin_amdgcn_wmma_f32_16x16x4_f32(
        /*neg_a=*/false, a, /*neg_b=*/false, b,
        /*c_mod=*/(short)0, c, /*reuse_a=*/false, /*reuse_b=*/false);
}

__device__ __forceinline__ v2f frag_a(const float* A, int lda, int lane, int k0) {
    int row = lane & 15;
    int kk  = k0 + ((lane >> 4) << 1);
    v2f a;
    a.x = A[row * lda + kk];
    a.y = A[row * lda + kk + 1];
    return a;
}

__device__ __forceinline__ v2f frag_b(const float* B, int ldb, int lane, int k0, int col0) {
    int col = col0 + (lane & 15);
    int kk  = k0 + ((lane >> 4) << 1);
    v2f b;
    b.x = B[kk * ldb + col];
    b.y = B[(kk + 1) * ldb + col];
    return b;
}

// ---------------------------------------------------------------------------
// Encoder: h = relu(x @ W1 + b1) @ W2 + b2      [N,64]->[N,128]
// One block = 16 rows; 8 waves = 8 column tiles of 128.
// ---------------------------------------------------------------------------
__global__ __launch_bounds__(256) void encoder_kernel(
    const float* __restrict__ x, const float* __restrict__ w1, const float* __restrict__ b1,
    const float* __restrict__ w2, const float* __restrict__ b2, float* __restrict__ h) {
    __shared__ float sA[16 * IN_DIM];   // x tile
    __shared__ float sT[16 * HIDDEN];   // relu(x@W1+b1) tile
    const int row0 = blockIdx.x * 16;
    const int tid  = threadIdx.x;
    const int wave = tid >> 5, lane = tid & 31;

    for (int i = tid; i < 16 * IN_DIM; i += 256) {
        int r = i >> 6, c = i & 63;
        sA[i] = x[(row0 + r) * IN_DIM + c];
    }
    __syncthreads();

    const int col0 = wave * 16;
    v8f acc = {};
#pragma unroll
    for (int k = 0; k < IN_DIM; k += 4)
        acc = wmma_k4(frag_a(sA, IN_DIM, lane, k), frag_b(w1, HIDDEN, lane, k, col0), acc);
    {
        int col = col0 + (lane & 15), rbase = (lane >> 4) << 3;
        float bias = b1[col];
#pragma unroll
        for (int i = 0; i < 8; ++i)
            sT[(rbase + i) * HIDDEN + col] = fmaxf(acc[i] + bias, 0.f);
    }
    __syncthreads();

    v8f acc2 = {};
#pragma unroll
    for (int k = 0; k < HIDDEN; k += 4)
        acc2 = wmma_k4(frag_a(sT, HIDDEN, lane, k), frag_b(w2, HIDDEN, lane, k, col0), acc2);
    {
        int col = col0 + (lane & 15), rbase = (lane >> 4) << 3;
        float bias = b2[col];
#pragma unroll
        for (int i = 0; i < 8; ++i)
            h[(row0 + rbase + i) * HIDDEN + col] = acc2[i] + bias;
    }
}

// ---------------------------------------------------------------------------
// GINEConv edge pass: msg = relu(h[src] + edge_attr @ Ew + Eb); agg[dst] += msg
// Block = 16 edges; 8 waves cover the 128 output columns; K=8 -> 2 WMMA steps.
// ---------------------------------------------------------------------------
__global__ __launch_bounds__(256) void edge_kernel(
    const float* __restrict__ h, const int* __restrict__ src, const int* __restrict__ dst,
    const float* __restrict__ eattr, const float* __restrict__ ew, const float* __restrict__ eb,
    float* __restrict__ agg) {
    __shared__ float sE[16 * EDGE_DIM];
    __shared__ int sSrc[16], sDst[16];
    const int e0  = blockIdx.x * 16;
    const int tid = threadIdx.x;
    const int wave = tid >> 5, lane = tid & 31;

    if (tid < 16) {
        sSrc[tid] = src[e0 + tid];
        sDst[tid] = dst[e0 + tid];
    }
    if (tid < 16 * EDGE_DIM) sE[tid] = eattr[e0 * EDGE_DIM + tid];
    __syncthreads();

    const int col0 = wave * 16;
    v8f acc = {};
#pragma unroll
    for (int k = 0; k < EDGE_DIM; k += 4)
        acc = wmma_k4(frag_a(sE, EDGE_DIM, lane, k), frag_b(ew, HIDDEN, lane, k, col0), acc);

    const int col = col0 + (lane & 15), rbase = (lane >> 4) << 3;
    const float bias = eb[col];
#pragma unroll
    for (int i = 0; i < 8; ++i) {
        int er = rbase + i;
        float v = fmaxf(acc[i] + bias + h[sSrc[er] * HIDDEN + col], 0.f);
        atomicAdd(&agg[sDst[er] * HIDDEN + col], v);
    }
}

// ---------------------------------------------------------------------------
// Node MLP: z = relu((h+agg)@W1+b1)@W2+b2; z written into agg buffer;
// per-column partial sums accumulated into colsum for GraphNorm mean.
// ---------------------------------------------------------------------------
__global__ __launch_bounds__(256) void node_mlp_kernel(
    const float* __restrict__ h, float* __restrict__ aggz,
    const float* __restrict__ w1, const float* __restrict__ b1,
    const float* __restrict__ w2, const float* __restrict__ b2,
    float* __restrict__ colsum) {
    __shared__ float sA[16 * HIDDEN];
    __shared__ float sT[16 * HIDDEN];
    const int row0 = blockIdx.x * 16;
    const int tid  = threadIdx.x;
    const int wave = tid >> 5, lane = tid & 31;

    for (int i = tid; i < 16 * HIDDEN; i += 256) {
        int r = i >> 7, c = i & 127;
        int g = (row0 + r) * HIDDEN + c;
        sA[i] = h[g] + aggz[g];
    }
    __syncthreads();

    const int col0 = wave * 16;
    v8f acc = {};
#pragma unroll
    for (int k = 0; k < HIDDEN; k += 4)
        acc = wmma_k4(frag_a(sA, HIDDEN, lane, k), frag_b(w1, HIDDEN, lane, k, col0), acc);
    {
        int col = col0 + (lane & 15), rbase = (lane >> 4) << 3;
        float bias = b1[col];
#pragma unroll
        for (int i = 0; i < 8; ++i)
            sT[(rbase + i) * HIDDEN + col] = fmaxf(acc[i] + bias, 0.f);
    }
    __syncthreads();

    v8f acc2 = {};
#pragma unroll
    for (int k = 0; k < HIDDEN; k += 4)
        acc2 = wmma_k4(frag_a(sT, HIDDEN, lane, k), frag_b(w2, HIDDEN, lane, k, col0), acc2);

    const int col = col0 + (lane & 15), rbase = (lane >> 4) << 3;
    const float bias = b2[col];
    float s = 0.f;
#pragma unroll
    for (int i = 0; i < 8; ++i) {
        float v = acc2[i] + bias;
        aggz[(row0 + rbase + i) * HIDDEN + col] = v;   // z overwrites agg (rows block-exclusive)
        s += v;
    }
    s += __shfl_xor(s, 16, 32);          // combine rows 0-7 with rows 8-15 (same column)
    if (lane < 16) atomicAdd(&colsum[col], s);
}

// ---------------------------------------------------------------------------
// GraphNorm stats + finalize
// ---------------------------------------------------------------------------
__global__ void mean_kernel(const float* __restrict__ colsum, float* __restrict__ mean) {
    int c = threadIdx.x;
    mean[c] = colsum[c] * (1.0f / (float)N_NODES);
}

__global__ __launch_bounds__(256) void sqsum_kernel(
    const float* __restrict__ z, const float* __restrict__ mean,
    const float* __restrict__ alpha, float* __restrict__ sqsum) {
    __shared__ float red[256];
    const int row0 = blockIdx.x * 16;
    const int tid  = threadIdx.x;
    const int col  = tid & 127;
    const int rsel = tid >> 7;           // 0: rows 0-7, 1: rows 8-15
    const float am = alpha[col] * mean[col];
    float s = 0.f;
#pragma unroll
    for (int i = 0; i < 8; ++i) {
        float o = z[(row0 + rsel * 8 + i) * HIDDEN + col] - am;
        s += o * o;
    }
    red[tid] = s;
    __syncthreads();
    if (tid < 128) atomicAdd(&sqsum[tid], red[tid] + red[tid + 128]);
}

__global__ __launch_bounds__(256) void finalize_kernel(
    const float* __restrict__ z, const float* __restrict__ mean, const float* __restrict__ sqsum,
    const float* __restrict__ alpha, const float* __restrict__ nw, const float* __restrict__ nb,
    float* __restrict__ h) {
    int i = blockIdx.x * 256 + threadIdx.x;
    if (i < N_NODES * HIDDEN) {
        int col = i & 127;
        float rstd = rsqrtf(sqsum[col] * (1.0f / (float)N_NODES) + EPS_NORM);
        float out  = z[i] - alpha[col] * mean[col];
        h[i] = fmaxf(nw[col] * out * rstd + nb[col], 0.f);
    }
}

// ---------------------------------------------------------------------------
// Decoder: out[n] = h[n,:] . dec_w + dec_b
// ---------------------------------------------------------------------------
__global__ __launch_bounds__(256) void decoder_kernel(
    const float* __restrict__ h, const float* __restrict__ dw, const float* __restrict__ db,
    float* __restrict__ out) {
    int n = blockIdx.x * 256 + threadIdx.x;
    if (n < N_NODES) {
        float s = db[0];
        const float* row = h + n * HIDDEN;
#pragma unroll 4
        for (int k = 0; k < HIDDEN; ++k) s += row[k] * dw[k];
        out[n] = s;
    }
}

// ---------------------------------------------------------------------------
extern "C" void kernel_launch(void* const* d_in, const int* in_sizes, int n_in,
                              void* d_out, int out_size, void* d_ws, size_t ws_size,
                              hipStream_t stream) {
    const float* x      = (const float*)d_in[0];
    const int*   eidx   = (const int*)d_in[1];
    const float* eattr  = (const float*)d_in[2];
    const float* enc_w1 = (const float*)d_in[3];
    const float* enc_b1 = (const float*)d_in[4];
    const float* enc_w2 = (const float*)d_in[5];
    const float* enc_b2 = (const float*)d_in[6];
    const float* edge_w = (const float*)d_in[7];   // [L,8,128]
    const float* edge_b = (const float*)d_in[8];   // [L,128]
    const float* mlp_w1 = (const float*)d_in[9];   // [L,128,128]
    const float* mlp_b1 = (const float*)d_in[10];
    const float* mlp_w2 = (const float*)d_in[11];
    const float* mlp_b2 = (const float*)d_in[12];
    const float* norm_w = (const float*)d_in[13];
    const float* norm_b = (const float*)d_in[14];
    const float* norm_a = (const float*)d_in[15];
    const float* dec_w  = (const float*)d_in[16];
    const float* dec_b  = (const float*)d_in[17];

    const int* src = eidx;
    const int* dst = eidx + E_EDGES;

    float* h      = (float*)d_ws;                         // [N,128]
    float* aggz   = h + (size_t)N_NODES * HIDDEN;         // [N,128] agg, then z
    float* colsum = aggz + (size_t)N_NODES * HIDDEN;      // [128]
    float* sqsum  = colsum + HIDDEN;                      // [128]
    float* meanv  = sqsum + HIDDEN;                       // [128]

    const int rowBlocks  = N_NODES / 16;   // 3125
    const int edgeBlocks = E_EDGES / 16;   // 37500

    encoder_kernel<<<rowBlocks, 256, 0, stream>>>(x, enc_w1, enc_b1, enc_w2, enc_b2, h);

    for (int l = 0; l < LAYERS; ++l) {
        hipMemsetAsync(aggz, 0, (size_t)N_NODES * HIDDEN * sizeof(float), stream);
        hipMemsetAsync(colsum, 0, 2 * HIDDEN * sizeof(float), stream);  // colsum + sqsum

        edge_kernel<<<edgeBlocks, 256, 0, stream>>>(
            h, src, dst, eattr,
            edge_w + (size_t)l * EDGE_DIM * HIDDEN, edge_b + (size_t)l * HIDDEN, aggz);

        node_mlp_kernel<<<rowBlocks, 256, 0, stream>>>(
            h, aggz,
            mlp_w1 + (size_t)l * HIDDEN * HIDDEN, mlp_b1 + (size_t)l * HIDDEN,
            mlp_w2 + (size_t)l * HIDDEN * HIDDEN, mlp_b2 + (size_t)l * HIDDEN, colsum);

        mean_kernel<<<1, HIDDEN, 0, stream>>>(colsum, meanv);

        sqsum_kernel<<<rowBlocks, 256, 0, stream>>>(
            aggz, meanv, norm_a + (size_t)l * HIDDEN, sqsum);

        finalize_kernel<<<(N_NODES * HIDDEN + 255) / 256, 256, 0, stream>>>(
            aggz, meanv, sqsum,
            norm_a + (size_t)l * HIDDEN, norm_w + (size_t)l * HIDDEN,
            norm_b + (size_t)l * HIDDEN, h);
    }

    decoder_kernel<<<(N_NODES + 255) / 256, 256, 0, stream>>>(h, dec_w, dec_b, (float*)d_out);
}